// NTMWriteHead_91044716740764
// MI455X (gfx1250) — compile-verified
//
#include <hip/hip_runtime.h>
#include <math.h>

// Problem constants (from reference)
#define Bx 64
#define Nx 16384
#define Mx 64
#define Cx 256
#define NOx 66          // M + 2
#define GAMMA 0.95f
#define EPSx 1e-16f
#define ROWS_PER_BLOCK 512

typedef float v2f __attribute__((ext_vector_type(2)));
typedef float v4f __attribute__((ext_vector_type(4)));
typedef float v8f __attribute__((ext_vector_type(8)));

// ---------------------------------------------------------------------------
// Kernel A: o = emb(64x256) @ W^T(256x66) + b  via V_WMMA_F32_16X16X4_F32
// Grid (4,5) tiles of 16x16, one wave32 per tile, 64 K-steps of 4.
// The out-of-range B columns (66..79 in the last tile) are handled WITHOUT
// exec-mask branches: clamped address + arithmetic zeroing, so the K-loop is
// branch-free (pure load + wmma) and EXEC stays all-1s as WMMA requires.
//
// f32 WMMA fragment layouts (ISA 7.12.2):
//   A 16x4:  lane l (row = l&15): VGPR0 = A[row][K0 + 2*(l>>4)], VGPR1 = +1
//   B 4x16:  lane l (col = l&15): VGPR0 = B[K0 + 2*(l>>4)][col], VGPR1 = +1
//   C/D 16x16: vgpr r, lane l -> row = r + 8*(l>>4), col = l&15
// ---------------------------------------------------------------------------
__global__ __launch_bounds__(32) void proj_wmma_kernel(
    const float* __restrict__ emb, const float* __restrict__ W,
    const float* __restrict__ bias,
    float* __restrict__ kmat, float* __restrict__ betav, float* __restrict__ gv)
{
    const int tm = blockIdx.x;        // row tile 0..3
    const int tn = blockIdx.y;        // col tile 0..4
    const int l  = threadIdx.x;       // lane 0..31
    const int q  = l & 15;
    const int h  = l >> 4;

    const int  arow   = tm * 16 + q;            // A row this lane feeds
    const int  bcol   = tn * 16 + q;            // B col this lane feeds
    const bool bvalid = (bcol < NOx);
    const int  bcol_c = bvalid ? bcol : 0;      // clamped: load always legal
    const float bscale = bvalid ? 1.f : 0.f;    // zero out invalid lanes

    const float* __restrict__ arow_p = emb + arow * Cx + 2 * h;
    const float* __restrict__ brow_p = W   + bcol_c * Cx + 2 * h;

    v8f acc = {};
    #pragma unroll 4
    for (int kb = 0; kb < Cx; kb += 4) {
        v2f a = *(const v2f*)(arow_p + kb);
        v2f w2 = *(const v2f*)(brow_p + kb);
        v2f bb;
        bb.x = w2.x * bscale;
        bb.y = w2.y * bscale;
        // (neg_a, A, neg_b, B, c_mod, C, reuse_a, reuse_b)
        acc = __builtin_amdgcn_wmma_f32_16x16x4_f32(
            false, a, false, bb, (short)0, acc, false, false);
    }

    if (!bvalid) return;                        // after all WMMAs
    const float bv = bias[bcol];
    #pragma unroll
    for (int r = 0; r < 8; ++r) {
        const int row = tm * 16 + r + 8 * h;
        const float val = acc[r] + bv;
        if (bcol < Mx) {
            kmat[row * Mx + bcol] = val;                             // k
        } else if (bcol == Mx) {
            betav[row] = (val > 20.f) ? val : log1pf(__expf(val));   // softplus
        } else {                                                     // col 65
            gv[row] = 1.f / (1.f + __expf(-val));                    // sigmoid
        }
    }
}

// ---------------------------------------------------------------------------
// Kernel A2: knorm[b] = ||k[b]||  (64 blocks x 64 threads)
// ---------------------------------------------------------------------------
__global__ __launch_bounds__(64) void knorm_kernel(
    const float* __restrict__ kmat, float* __restrict__ knormv)
{
    const int b = blockIdx.x;
    const int t = threadIdx.x;
    float v = kmat[b * Mx + t];
    v *= v;
    #pragma unroll
    for (int mm = 16; mm >= 1; mm >>= 1) v += __shfl_xor(v, mm, 32);
    __shared__ float sm[2];
    if ((t & 31) == 0) sm[t >> 5] = v;
    __syncthreads();
    if (t == 0) knormv[b] = sqrtf(sm[0] + sm[1]);
}

// ---------------------------------------------------------------------------
// Kernel B: the 512 MB streaming pass (bandwidth-critical).
// Per memory row (64 f32 = 256 B): 16 lanes x float4 NT load, compute
// dot(k,row) and ||row||^2 via shfl-xor reduce, emit sim & w_w, and write
// new_memory = memory + w_w * k with NT stores (keep the 512 MB stream out
// of the 192 MB L2; the 8 MB sim/w_w working set stays cached instead).
// Block = 256 thr (8 waves); each wave does 2 rows/iter (half-wave per row).
// ---------------------------------------------------------------------------
__global__ __launch_bounds__(256) void stream_kernel(
    const float* __restrict__ memory, const float* __restrict__ w_prev,
    const float* __restrict__ kmat,  const float* __restrict__ knormv,
    const float* __restrict__ gv,
    float* __restrict__ newmem, float* __restrict__ simv, float* __restrict__ wwv)
{
    const int b = blockIdx.y;
    const int w = threadIdx.x >> 5;
    const int l = threadIdx.x & 31;
    const int q = l & 15;
    const int h = l >> 4;

    const v4f  kq    = *(const v4f*)(kmat + b * Mx + q * 4);
    const float knrm = knormv[b];
    const float g    = gv[b];
    const float* wr_prev  = w_prev + (size_t)b * 3 * Nx + Nx;
    const float* wlu_prev = wr_prev + Nx;

    const int base = blockIdx.x * ROWS_PER_BLOCK + w * 2 + h;
    #pragma unroll 2
    for (int it = 0; it < ROWS_PER_BLOCK / 16; ++it) {
        const int    row = base + it * 16;
        const size_t off = ((size_t)b * Nx + row) * Mx + q * 4;

        v4f m4 = __builtin_nontemporal_load((const v4f*)(memory + off));
        float dot = m4.x * kq.x + m4.y * kq.y + m4.z * kq.z + m4.w * kq.w;
        float nrm = m4.x * m4.x + m4.y * m4.y + m4.z * m4.z + m4.w * m4.w;
        #pragma unroll
        for (int mm = 8; mm >= 1; mm >>= 1) {      // reduce within 16-lane group
            dot += __shfl_xor(dot, mm, 32);
            nrm += __shfl_xor(nrm, mm, 32);
        }

        const float wrp  = wr_prev[row];
        const float wlup = wlu_prev[row];
        const float ww   = g * wrp + (1.f - g) * wlup;

        v4f nm;
        nm.x = m4.x + ww * kq.x;  nm.y = m4.y + ww * kq.y;
        nm.z = m4.z + ww * kq.z;  nm.w = m4.w + ww * kq.w;
        __builtin_nontemporal_store(nm, (v4f*)(newmem + off));

        if (q == 0) {
            simv[(size_t)b * Nx + row] = dot / (knrm * sqrtf(nrm) + EPSx);
            wwv [(size_t)b * Nx + row] = ww;
        }
    }
}

// ---------------------------------------------------------------------------
// Kernel C: per-batch softmax over N=16384, w_u, bottom-4 -> w_lu one-hot.
// 64 blocks x 1024 threads (32 waves); each thread owns 16 elements
// (idx = i*1024 + t).
// ---------------------------------------------------------------------------
__device__ __forceinline__ float block_reduce_max(float v, float* sm) {
    #pragma unroll
    for (int mm = 16; mm >= 1; mm >>= 1) v = fmaxf(v, __shfl_xor(v, mm, 32));
    if ((threadIdx.x & 31) == 0) sm[threadIdx.x >> 5] = v;
    __syncthreads();
    if (threadIdx.x < 32) {
        v = sm[threadIdx.x];
        #pragma unroll
        for (int mm = 16; mm >= 1; mm >>= 1) v = fmaxf(v, __shfl_xor(v, mm, 32));
        if (threadIdx.x == 0) sm[0] = v;
    }
    __syncthreads();
    v = sm[0];
    __syncthreads();
    return v;
}

__device__ __forceinline__ float block_reduce_sum(float v, float* sm) {
    #pragma unroll
    for (int mm = 16; mm >= 1; mm >>= 1) v += __shfl_xor(v, mm, 32);
    if ((threadIdx.x & 31) == 0) sm[threadIdx.x >> 5] = v;
    __syncthreads();
    if (threadIdx.x < 32) {
        v = sm[threadIdx.x];
        #pragma unroll
        for (int mm = 16; mm >= 1; mm >>= 1) v += __shfl_xor(v, mm, 32);
        if (threadIdx.x == 0) sm[0] = v;
    }
    __syncthreads();
    v = sm[0];
    __syncthreads();
    return v;
}

__global__ __launch_bounds__(1024) void finalize_kernel(
    const float* __restrict__ simv, const float* __restrict__ wwv,
    const float* __restrict__ w_prev, const float* __restrict__ betav,
    float* __restrict__ wout)
{
    const int b = blockIdx.x;
    const int t = threadIdx.x;
    __shared__ float redv[32];
    __shared__ int   redi[32];
    __shared__ float bcv;
    __shared__ int   bci;

    const float beta = betav[b];
    const float* sim_b   = simv   + (size_t)b * Nx;
    const float* ww_b    = wwv    + (size_t)b * Nx;
    const float* wu_prev = w_prev + (size_t)b * 3 * Nx;
    float* wu_out  = wout + (size_t)b * 3 * Nx;
    float* wr_out  = wu_out + Nx;
    float* wlu_out = wr_out + Nx;

    float s[16];
    #pragma unroll
    for (int i = 0; i < 16; ++i) s[i] = beta * sim_b[i * 1024 + t];

    float mx = -3.4e38f;
    #pragma unroll
    for (int i = 0; i < 16; ++i) mx = fmaxf(mx, s[i]);
    mx = block_reduce_max(mx, redv);

    float sum = 0.f;
    #pragma unroll
    for (int i = 0; i < 16; ++i) { s[i] = __expf(s[i] - mx); sum += s[i]; }
    sum = block_reduce_sum(sum, redv);
    const float inv = 1.f / sum;

    float wu[16];
    #pragma unroll
    for (int i = 0; i < 16; ++i) {
        const int idx = i * 1024 + t;
        const float wr = s[i] * inv;
        const float u  = GAMMA * wu_prev[idx] + wr + ww_b[idx];
        wu[i] = u;
        wr_out[idx] = wr;
        wu_out[idx] = u;
    }

    // 4 rounds of block-argmin over w_u (ties -> lower index, like lax.top_k)
    int sel[4];
    #pragma unroll
    for (int round = 0; round < 4; ++round) {
        float bv = 3.4e38f; int bi = 0x40000000;
        #pragma unroll
        for (int i = 0; i < 16; ++i) {
            const int idx = i * 1024 + t;
            if (wu[i] < bv || (wu[i] == bv && idx < bi)) { bv = wu[i]; bi = idx; }
        }
        #pragma unroll
        for (int mm = 16; mm >= 1; mm >>= 1) {
            const float ov = __shfl_xor(bv, mm, 32);
            const int   oi = __shfl_xor(bi, mm, 32);
            if (ov < bv || (ov == bv && oi < bi)) { bv = ov; bi = oi; }
        }
        if ((t & 31) == 0) { redv[t >> 5] = bv; redi[t >> 5] = bi; }
        __syncthreads();
        if (t < 32) {
            bv = redv[t]; bi = redi[t];
            #pragma unroll
            for (int mm = 16; mm >= 1; mm >>= 1) {
                const float ov = __shfl_xor(bv, mm, 32);
                const int   oi = __shfl_xor(bi, mm, 32);
                if (ov < bv || (ov == bv && oi < bi)) { bv = ov; bi = oi; }
            }
            if (t == 0) { bcv = bv; bci = bi; }
        }
        __syncthreads();
        const int widx = bci;
        sel[round] = widx;
        if ((widx & 1023) == t) wu[widx >> 10] = 3.4e38f;  // mask out winner
        __syncthreads();
    }

    #pragma unroll
    for (int i = 0; i < 16; ++i) {
        const int idx = i * 1024 + t;
        const float v = (idx == sel[0] || idx == sel[1] || idx == sel[2] || idx == sel[3])
                        ? 1.f : 0.f;
        wlu_out[idx] = v;
    }
}

// ---------------------------------------------------------------------------
extern "C" void kernel_launch(void* const* d_in, const int* in_sizes, int n_in,
                              void* d_out, int out_size, void* d_ws, size_t ws_size,
                              hipStream_t stream)
{
    const float* emb    = (const float*)d_in[0];   // (64,256)
    const float* w_prev = (const float*)d_in[1];   // (64,3,16384)
    const float* memory = (const float*)d_in[2];   // (64,16384,64)
    const float* W      = (const float*)d_in[3];   // (66,256)
    const float* bias   = (const float*)d_in[4];   // (66,)
    // d_in[5] = n (always 4)

    float* ws    = (float*)d_ws;
    float* kmat  = ws;                      // 64*64
    float* knorm = ws + 4096;               // 64
    float* betav = ws + 4160;               // 64
    float* gv    = ws + 4224;               // 64
    float* simv  = ws + 4288;               // 64*16384
    float* wwv   = simv + (size_t)Bx * Nx;  // 64*16384

    float* wout   = (float*)d_out;                       // (64,3,16384)
    float* newmem = wout + (size_t)3 * Bx * Nx;          // (64,16384,64)

    proj_wmma_kernel<<<dim3(4, 5), 32, 0, stream>>>(emb, W, bias, kmat, betav, gv);
    knorm_kernel<<<Bx, 64, 0, stream>>>(kmat, knorm);
    stream_kernel<<<dim3(Nx / ROWS_PER_BLOCK, Bx), 256, 0, stream>>>(
        memory, w_prev, kmat, knorm, gv, newmem, simv, wwv);
    finalize_kernel<<<Bx, 1024, 0, stream>>>(simv, wwv, w_prev, betav, wout);
}